// SubjectLayers_49606872268952
// MI455X (gfx1250) — compile-verified
//
#include <hip/hip_runtime.h>

// Problem constants (from reference)
#define BB 4096
#define CC 256
#define DD 512
#define SS 8

// Workspace layout (ints)
#define MAX_TILES 264            // sum ceil(cnt_s/16) <= B/16 + S = 264
#define WS_NTILES 0
#define WS_DESC   16             // 3 ints per tile
#define WS_ROWIDX (16 + 3 * MAX_TILES)

typedef __attribute__((ext_vector_type(2))) float v2f;
typedef __attribute__((ext_vector_type(8))) float v8f;

// ---------------------------------------------------------------------------
// Kernel 1: bucket rows by subject, build tile descriptors.
// Single block, 256 threads. Deterministic output values (row order within a
// bucket does not affect any output element).
// ---------------------------------------------------------------------------
__global__ __launch_bounds__(256) void subject_bucket_kernel(
    const int* __restrict__ subjects, int* __restrict__ ws) {
    __shared__ int cnt[SS];
    __shared__ int off[SS];
    __shared__ int cur[SS];
    const int t = threadIdx.x;

    if (t < SS) cnt[t] = 0;
    __syncthreads();

    for (int b = t; b < BB; b += 256) atomicAdd(&cnt[subjects[b]], 1);
    __syncthreads();

    if (t == 0) {
        int run = 0, nt = 0;
        for (int s = 0; s < SS; ++s) {
            off[s] = run;
            const int c = cnt[s];
            for (int tb = 0; tb < c; tb += 16) {
                ws[WS_DESC + 3 * nt + 0] = s;
                ws[WS_DESC + 3 * nt + 1] = run + tb;
                ws[WS_DESC + 3 * nt + 2] = (c - tb < 16) ? (c - tb) : 16;
                ++nt;
            }
            run += c;
        }
        ws[WS_NTILES] = nt;
    }
    __syncthreads();

    if (t < SS) cur[t] = off[t];
    __syncthreads();

    for (int b = t; b < BB; b += 256) {
        const int s   = subjects[b];
        const int pos = atomicAdd(&cur[s], 1);
        ws[WS_ROWIDX + pos] = b;
    }
}

// ---------------------------------------------------------------------------
// Kernel 2: grouped GEMM tile. One block = one 16-row tile x full D=512.
// 8 waves; each wave computes four 16x16 output tiles via
// v_wmma_f32_16x16x4_f32 over K=256 in steps of 4.
// ---------------------------------------------------------------------------
#define A_STRIDE 260  // 16x256 A tile padded to avoid LDS bank conflicts

__global__ __launch_bounds__(256) void subject_gemm_kernel(
    const float* __restrict__ x,        // (B,C)
    const float* __restrict__ weights,  // (S,C,D)
    const float* __restrict__ bias,     // (S,D)
    const int* __restrict__ ws,
    float* __restrict__ out) {          // (B,D)
    __shared__ float At[16 * A_STRIDE];
    __shared__ int   srow[16];
    __shared__ int   sdesc[3];

    const int tid = threadIdx.x;

    const int ntiles = ws[WS_NTILES];
    if ((int)blockIdx.x >= ntiles) return;   // wave-uniform: EXEC stays full

    if (tid < 3) sdesc[tid] = ws[WS_DESC + 3 * blockIdx.x + tid];
    __syncthreads();
    const int s     = sdesc[0];
    const int base  = sdesc[1];
    const int valid = sdesc[2];

    if (tid < 16) {
        const int r = (tid < valid) ? tid : (valid - 1);  // clamp tail rows
        srow[tid] = ws[WS_ROWIDX + base + r];
    }
    __syncthreads();

    // Stage the 16xC A tile into LDS (coalesced: 16 passes of 256 floats).
    for (int i = 0; i < 16; ++i) {
        At[i * A_STRIDE + tid] = x[(long)srow[i] * CC + tid];
    }
    __syncthreads();

    const int lane = tid & 31;
    const int wave = tid >> 5;
    const int half = lane >> 4;   // 0: K even-pair low, 1: K even-pair high
    const int m    = lane & 15;

    const float* __restrict__ W = weights + (long)s * (CC * DD);
    const int nb = wave * 64 + m;  // base column for this lane's 4 N-tiles

    v8f acc0 = {}, acc1 = {}, acc2 = {}, acc3 = {};

#pragma unroll 4
    for (int kb = 0; kb < CC; kb += 4) {
        const int k0 = kb + 2 * half;
        // A fragment: lane m holds A[m][k0], A[m][k0+1]  (16x4 f32 layout)
        const v2f a = *(const v2f*)&At[m * A_STRIDE + k0];
        // B fragments: lane n=m holds B[k0][n], B[k0+1][n]  (4x16 layout)
        const float* __restrict__ wk = W + (long)k0 * DD;
        const v2f b0 = { wk[nb],       wk[DD + nb]       };
        const v2f b1 = { wk[nb + 16],  wk[DD + nb + 16]  };
        const v2f b2 = { wk[nb + 32],  wk[DD + nb + 32]  };
        const v2f b3 = { wk[nb + 48],  wk[DD + nb + 48]  };

        acc0 = __builtin_amdgcn_wmma_f32_16x16x4_f32(false, a, false, b0,
                                                     (short)0, acc0, false, false);
        acc1 = __builtin_amdgcn_wmma_f32_16x16x4_f32(false, a, false, b1,
                                                     (short)0, acc1, false, false);
        acc2 = __builtin_amdgcn_wmma_f32_16x16x4_f32(false, a, false, b2,
                                                     (short)0, acc2, false, false);
        acc3 = __builtin_amdgcn_wmma_f32_16x16x4_f32(false, a, false, b3,
                                                     (short)0, acc3, false, false);
    }

    // Epilogue: add bias, scatter rows back to their sample indices.
    const float* __restrict__ bs = bias + (long)s * DD;
    const float bv0 = bs[nb];
    const float bv1 = bs[nb + 16];
    const float bv2 = bs[nb + 32];
    const float bv3 = bs[nb + 48];

#pragma unroll
    for (int r = 0; r < 8; ++r) {
        const int rl = r + 8 * half;  // local row in the 16-row tile
        if (rl < valid) {
            float* __restrict__ orow = out + (long)srow[rl] * DD;
            orow[nb]      = acc0[r] + bv0;
            orow[nb + 16] = acc1[r] + bv1;
            orow[nb + 32] = acc2[r] + bv2;
            orow[nb + 48] = acc3[r] + bv3;
        }
    }
}

// ---------------------------------------------------------------------------
extern "C" void kernel_launch(void* const* d_in, const int* in_sizes, int n_in,
                              void* d_out, int out_size, void* d_ws, size_t ws_size,
                              hipStream_t stream) {
    const float* x        = (const float*)d_in[0];
    const int*   subjects = (const int*)d_in[1];
    const float* weights  = (const float*)d_in[2];
    const float* bias     = (const float*)d_in[3];
    float*       out      = (float*)d_out;
    int*         ws       = (int*)d_ws;

    subject_bucket_kernel<<<1, 256, 0, stream>>>(subjects, ws);
    subject_gemm_kernel<<<MAX_TILES, 256, 0, stream>>>(x, weights, bias, ws, out);
}